// ADCLMBRec_49804440764586
// MI455X (gfx1250) — compile-verified
//
#include <hip/hip_runtime.h>

// ---------------------------------------------------------------------------
// ADCLMBRec for MI455X (gfx1250, wave32).
// Memory-bound edge scatter (atomics into L2-resident prop), WMMA f32 16x16x4
// for the (prop*rela)@W_gc GEMMs, wave-per-node VALU for 3x3 attention.
// ---------------------------------------------------------------------------

#define N_USERS 70000
#define N_ITEMS 30000
#define N_NODES 100000   // N_USERS + N_ITEMS
#define R_REL   3
#define D_DIM   64
#define L_LAYERS 2
#define E_EDGES 1600000

typedef float v2f __attribute__((ext_vector_type(2)));
typedef float v8f __attribute__((ext_vector_type(8)));

__device__ __forceinline__ float warp_sum32(float v) {
#pragma unroll
  for (int m = 16; m >= 1; m >>= 1) v += __shfl_xor(v, m, 32);
  return v;
}

// ego/all_emb <- broadcast concat(user_emb, item_emb) over R
__global__ __launch_bounds__(256) void k_init(const float* __restrict__ ue,
                                              const float* __restrict__ ie,
                                              float* __restrict__ ego,
                                              float* __restrict__ all_emb) {
  long long idx = (long long)blockIdx.x * 256 + threadIdx.x;
  if (idx >= (long long)N_NODES * D_DIM) return;
  int n = (int)(idx / D_DIM), d = (int)(idx % D_DIM);
  float v = (n < N_USERS) ? ue[(long long)n * D_DIM + d]
                          : ie[(long long)(n - N_USERS) * D_DIM + d];
#pragma unroll
  for (int r = 0; r < R_REL; ++r) {
    long long o = ((long long)n * R_REL + r) * D_DIM + d;
    ego[o] = v;
    all_emb[o] = v;
  }
}

__global__ __launch_bounds__(256) void k_zero(float* __restrict__ p, long long count) {
  long long i = (long long)blockIdx.x * 256 + threadIdx.x;
  if (i < count) p[i] = 0.f;
}

// prop[row, r, :] += ego[col, r, :] * val   (16 threads/edge, float4 gathers)
__global__ __launch_bounds__(256) void k_scatter(const float* __restrict__ ego,
                                                 const float* __restrict__ vals,
                                                 const int* __restrict__ rows,
                                                 const int* __restrict__ cols,
                                                 float* __restrict__ prop) {
  long long gt = (long long)blockIdx.x * 256 + threadIdx.x;
  long long e = gt >> 4;
  int q = (int)(gt & 15);
  if (e >= (long long)R_REL * E_EDGES) return;
  int r = (int)(e / E_EDGES);
  int col = cols[e];
  int row = rows[e];
  float val = vals[e];
  const float4 m = ((const float4*)(ego + ((long long)col * R_REL + r) * D_DIM))[q];
  float* dst = prop + ((long long)row * R_REL + r) * D_DIM + q * 4;
  atomicAdd(dst + 0, m.x * val);
  atomicAdd(dst + 1, m.y * val);
  atomicAdd(dst + 2, m.z * val);
  atomicAdd(dst + 3, m.w * val);
}

// est[n,r,:] = leaky_relu( (prop[n,r,:] * rela[r,:]) @ W )   [in-place est==prop OK]
// One wave -> 16x64 output tile via 64x V_WMMA_F32_16X16X4_F32.
__global__ __launch_bounds__(256) void k_gemm(const float* prop,
                                              const float* __restrict__ W,
                                              const float* __restrict__ rela,
                                              float* est) {
  __shared__ float lw[64 * 68];  // W[k][d] with padded stride 68 (bank-conflict free)
  const int r = blockIdx.y;
  const int tid = threadIdx.x;
  for (int i = tid; i < 4096; i += 256) lw[(i >> 6) * 68 + (i & 63)] = W[i];
  __syncthreads();

  const int wv = tid >> 5, lane = tid & 31;
  const int tile = blockIdx.x * 8 + wv;
  if (tile >= N_NODES / 16) return;

  const int half = lane >> 4;       // lanes 16-31 hold K+2 / M+8 parts
  const int m = lane & 15;
  const int row = tile * 16 + m;
  const float* arow = prop + ((long long)row * R_REL + r) * D_DIM;
  const float* rl = rela + r * D_DIM;

  // A fragments: 16 K-steps of 16x4 f32 (v2f per lane), rela folded in.
  v2f a[16];
#pragma unroll
  for (int kk = 0; kk < 16; ++kk) {
    int kb = kk * 4 + half * 2;
    a[kk].x = arow[kb] * rl[kb];
    a[kk].y = arow[kb + 1] * rl[kb + 1];
  }

  const int ncol = lane & 15;
#pragma unroll
  for (int nt = 0; nt < 4; ++nt) {
    v8f c = {0.f, 0.f, 0.f, 0.f, 0.f, 0.f, 0.f, 0.f};
#pragma unroll
    for (int kk = 0; kk < 16; ++kk) {
      int kb = kk * 4 + half * 2;
      v2f b;
      b.x = lw[kb * 68 + nt * 16 + ncol];
      b.y = lw[(kb + 1) * 68 + nt * 16 + ncol];
      c = __builtin_amdgcn_wmma_f32_16x16x4_f32(false, a[kk], false, b,
                                                (short)0, c, false, false);
    }
#pragma unroll
    for (int i = 0; i < 8; ++i) {
      float v = c[i];
      v = (v > 0.f) ? v : 0.01f * v;  // leaky_relu, slope 0.01
      est[((long long)(tile * 16 + i + half * 8) * R_REL + r) * D_DIM + nt * 16 + ncol] = v;
    }
  }
}

// Per-node 3x3 attention: ego = softmax(est est^T * scale) @ est; all_emb += ego.
__global__ __launch_bounds__(256) void k_attn(const float* est, float* ego,
                                              float* all_emb, float scale) {
  const int wv = threadIdx.x >> 5, lane = threadIdx.x & 31;
  const int n = blockIdx.x * 8 + wv;
  if (n >= N_NODES) return;
  const float* base = est + (long long)n * (R_REL * D_DIM);
  float e[3][2];
#pragma unroll
  for (int r = 0; r < 3; ++r) {
    e[r][0] = base[r * 64 + lane];
    e[r][1] = base[r * 64 + lane + 32];
  }
  float d00 = warp_sum32(e[0][0] * e[0][0] + e[0][1] * e[0][1]);
  float d01 = warp_sum32(e[0][0] * e[1][0] + e[0][1] * e[1][1]);
  float d02 = warp_sum32(e[0][0] * e[2][0] + e[0][1] * e[2][1]);
  float d11 = warp_sum32(e[1][0] * e[1][0] + e[1][1] * e[1][1]);
  float d12 = warp_sum32(e[1][0] * e[2][0] + e[1][1] * e[2][1]);
  float d22 = warp_sum32(e[2][0] * e[2][0] + e[2][1] * e[2][1]);
  float dot[3][3] = {{d00, d01, d02}, {d01, d11, d12}, {d02, d12, d22}};
  float att[3][3];
#pragma unroll
  for (int r = 0; r < 3; ++r) {
    float x0 = dot[r][0] * scale, x1 = dot[r][1] * scale, x2 = dot[r][2] * scale;
    float mx = fmaxf(x0, fmaxf(x1, x2));
    float s0 = expf(x0 - mx), s1 = expf(x1 - mx), s2 = expf(x2 - mx);
    float inv = 1.f / (s0 + s1 + s2);
    att[r][0] = s0 * inv; att[r][1] = s1 * inv; att[r][2] = s2 * inv;
  }
#pragma unroll
  for (int r = 0; r < 3; ++r) {
#pragma unroll
    for (int j = 0; j < 2; ++j) {
      float o = att[r][0] * e[0][j] + att[r][1] * e[1][j] + att[r][2] * e[2][j];
      long long idx = (long long)n * 192 + r * 64 + lane + j * 32;
      ego[idx] = o;
      all_emb[idx] += o;
    }
  }
}

// Final: att row 2 of softmax(all_emb all_emb^T); final = [a0, a1, mid2] / 3
__global__ __launch_bounds__(256) void k_final(const float* __restrict__ all_emb,
                                               float* __restrict__ u_out,
                                               float* __restrict__ i_out) {
  const int wv = threadIdx.x >> 5, lane = threadIdx.x & 31;
  const int n = blockIdx.x * 8 + wv;
  if (n >= N_NODES) return;
  const float* base = all_emb + (long long)n * 192;
  float a[3][2];
#pragma unroll
  for (int r = 0; r < 3; ++r) {
    a[r][0] = base[r * 64 + lane];
    a[r][1] = base[r * 64 + lane + 32];
  }
  float d20 = warp_sum32(a[2][0] * a[0][0] + a[2][1] * a[0][1]);
  float d21 = warp_sum32(a[2][0] * a[1][0] + a[2][1] * a[1][1]);
  float d22 = warp_sum32(a[2][0] * a[2][0] + a[2][1] * a[2][1]);
  float mx = fmaxf(d20, fmaxf(d21, d22));
  float s0 = expf(d20 - mx), s1 = expf(d21 - mx), s2 = expf(d22 - mx);
  float inv = 1.f / (s0 + s1 + s2);
  float t0 = s0 * inv, t1 = s1 * inv, t2 = s2 * inv;
  const float inv3 = 1.f / 3.f;
  float* dst = (n < N_USERS) ? (u_out + (long long)n * 192)
                             : (i_out + (long long)(n - N_USERS) * 192);
#pragma unroll
  for (int j = 0; j < 2; ++j) {
    float mid2 = t0 * a[0][j] + t1 * a[1][j] + t2 * a[2][j];
    dst[0 * 64 + lane + j * 32] = a[0][j] * inv3;
    dst[1 * 64 + lane + j * 32] = a[1][j] * inv3;
    dst[2 * 64 + lane + j * 32] = mid2 * inv3;
  }
}

__global__ void k_rela_init(const float* __restrict__ rel_emb,
                            float* __restrict__ rc, float* __restrict__ rs) {
  int t = threadIdx.x;
  if (t < 192) { rc[t] = rel_emb[t]; rs[t] = rel_emb[t]; }
}

// rela_cur = rela_cur @ W_rel[l]; rela_sum += rela_cur   (single block, 192 thr)
__global__ void k_rela_update(float* __restrict__ rc, float* __restrict__ rs,
                              const float* __restrict__ Wr) {
  __shared__ float cur[192];
  int t = threadIdx.x;
  cur[t] = rc[t];
  __syncthreads();
  int r = t >> 6, d = t & 63;
  float acc = 0.f;
#pragma unroll 8
  for (int k = 0; k < 64; ++k) acc += cur[r * 64 + k] * Wr[k * 64 + d];
  rc[t] = acc;
  rs[t] += acc;
}

// zero i_emb's appended row; rela_out = rela_sum / 3
__global__ void k_epilogue(const float* __restrict__ rs,
                           float* __restrict__ zero_row,
                           float* __restrict__ rela_out) {
  int t = threadIdx.x;
  if (t < 192) zero_row[t] = 0.f;
  else if (t < 384) rela_out[t - 192] = rs[t - 192] * (1.f / 3.f);
}

// Per-user GRU gate + scores (one wave per user).
__global__ __launch_bounds__(256) void k_gru(const float* __restrict__ u,
                                             const float* __restrict__ gw,
                                             const float* __restrict__ gb,
                                             const float* __restrict__ tra,
                                             float* __restrict__ s1,
                                             float* __restrict__ s2) {
  const int wv = threadIdx.x >> 5, lane = threadIdx.x & 31;
  const int n = blockIdx.x * 8 + wv;
  if (n >= N_USERS) return;
  const float* ub = u + (long long)n * 192;
  float uv[3][2], tmp[3][2];
#pragma unroll
  for (int r = 0; r < 3; ++r) {
    uv[r][0] = ub[r * 64 + lane];
    uv[r][1] = ub[r * 64 + lane + 32];
  }
#pragma unroll
  for (int r = 0; r < 3; ++r) {
    float acc0 = gb[r * 64 + lane];
    float acc1 = gb[r * 64 + lane + 32];
    const float* Wr = gw + r * 4096;
#pragma unroll 8
    for (int k = 0; k < 64; ++k) {
      float us = (k < 32) ? __shfl(uv[r][0], k, 32) : __shfl(uv[r][1], k - 32, 32);
      acc0 += us * Wr[k * 64 + lane];
      acc1 += us * Wr[k * 64 + lane + 32];
    }
    tmp[r][0] = acc0; tmp[r][1] = acc1;
  }
  float aux1_0 = uv[0][0] * tmp[0][0], aux1_1 = uv[0][1] * tmp[0][1];
  float aux2_0 = uv[1][0] * tmp[1][0], aux2_1 = uv[1][1] * tmp[1][1];
  float tgt_0  = uv[2][0] * tmp[2][0], tgt_1  = uv[2][1] * tmp[2][1];
  float p1 = tgt_0 * tra[lane] + tgt_1 * tra[lane + 32]
           + aux1_0 * tra[64 + lane] + aux1_1 * tra[64 + lane + 32];
  float p2 = tgt_0 * tra[128 + lane] + tgt_1 * tra[128 + lane + 32]
           + aux2_0 * tra[192 + lane] + aux2_1 * tra[192 + lane + 32];
  p1 = warp_sum32(p1);
  p2 = warp_sum32(p2);
  if (lane == 0) { s1[n] = p1; s2[n] = p2; }
}

extern "C" void kernel_launch(void* const* d_in, const int* in_sizes, int n_in,
                              void* d_out, int out_size, void* d_ws, size_t ws_size,
                              hipStream_t stream) {
  const float* user_emb = (const float*)d_in[0];
  const float* item_emb = (const float*)d_in[1];
  const float* rel_emb  = (const float*)d_in[2];
  const float* W_gc     = (const float*)d_in[3];
  const float* W_rel    = (const float*)d_in[4];
  const float* gru_w    = (const float*)d_in[5];
  const float* gru_b    = (const float*)d_in[6];
  const float* tra      = (const float*)d_in[7];
  const float* adj_vals = (const float*)d_in[8];
  const int*   adj_rows = (const int*)d_in[9];
  const int*   adj_cols = (const int*)d_in[10];

  const size_t NND = (size_t)N_NODES * R_REL * D_DIM;  // 19.2M floats
  float* ws       = (float*)d_ws;
  float* ego      = ws;
  float* all_emb  = ws + NND;
  float* prop     = ws + 2 * NND;  // reused in-place as est
  float* rela_cur = ws + 3 * NND;
  float* rela_sum = rela_cur + 192;

  float* out      = (float*)d_out;
  float* u_out    = out;
  float* i_out    = out + (size_t)N_USERS * 192;
  float* rela_out = i_out + (size_t)(N_ITEMS + 1) * 192;
  float* s1       = rela_out + 192;
  float* s2       = s1 + N_USERS;

  {
    long long tot = (long long)N_NODES * D_DIM;
    k_init<<<(unsigned)((tot + 255) / 256), 256, 0, stream>>>(user_emb, item_emb, ego, all_emb);
  }
  k_rela_init<<<1, 192, 0, stream>>>(rel_emb, rela_cur, rela_sum);

  const float inv_sqrt = 0.08838834764831845f;  // 1/sqrt(128)

  for (int l = 0; l < L_LAYERS; ++l) {
    k_zero<<<(unsigned)((NND + 255) / 256), 256, 0, stream>>>(prop, (long long)NND);
    {
      long long tot = (long long)R_REL * E_EDGES * 16;
      k_scatter<<<(unsigned)((tot + 255) / 256), 256, 0, stream>>>(ego, adj_vals, adj_rows,
                                                                   adj_cols, prop);
    }
    {
      dim3 g((N_NODES / 16 + 7) / 8, R_REL);
      k_gemm<<<g, 256, 0, stream>>>(prop, W_gc + (size_t)l * 4096, rela_cur, prop);
    }
    k_attn<<<(N_NODES + 7) / 8, 256, 0, stream>>>(prop, ego, all_emb, inv_sqrt);
    k_rela_update<<<1, 192, 0, stream>>>(rela_cur, rela_sum, W_rel + (size_t)l * 4096);
  }

  k_final<<<(N_NODES + 7) / 8, 256, 0, stream>>>(all_emb, u_out, i_out);
  k_epilogue<<<1, 384, 0, stream>>>(rela_sum, i_out + (size_t)N_ITEMS * 192, rela_out);
  k_gru<<<(N_USERS + 7) / 8, 256, 0, stream>>>(u_out, gru_w, gru_b, tra, s1, s2);
}